// RoIAlignAvg_30872224923785
// MI455X (gfx1250) — compile-verified
//
#include <hip/hip_runtime.h>

typedef __attribute__((ext_vector_type(2))) float v2f;
typedef __attribute__((ext_vector_type(4))) float v4f;
typedef __attribute__((ext_vector_type(8))) float v8f;

#define FH 100
#define FW 152
#define NCH 256
#define ROI_SCALE 0.0625f

// out(7x7) = Uh(7x16,pad16) * G(16x16) * Uw^T(16x7,pad16) per (roi, channel).
// G is the gathered 16x16 feature tile (row pairs hs[a],hs[a]+1; col pairs ws[b],ws[b]+1).
// Pooling (0.25 * 2x2 sum) and validity masking are folded into Uh/Uw.
__launch_bounds__(256)
__global__ void roi_align_wmma_kernel(const float* __restrict__ feat,
                                      const float* __restrict__ rois,
                                      const int*   __restrict__ bids,
                                      float* __restrict__ out)
{
    __shared__ float sUh[16][16];       // Uh zero-padded to 16x16
    __shared__ float sUw[16][16];       // Uw zero-padded to 16x16
    __shared__ int   sFr[16];           // gathered feature row index per p
    __shared__ int   sFc[16];           // gathered feature col index per q
    __shared__ __align__(16) float sT[8][256];  // per-wave D->A bounce buffer

    const int r   = blockIdx.x;
    const int tid = threadIdx.x;

    // zero-pad weight tables (256 threads cover 2x 16x16)
    sUh[tid >> 4][tid & 15] = 0.0f;
    sUw[tid >> 4][tid & 15] = 0.0f;
    __syncthreads();

    // Per-ROI setup: 32 lanes compute the 16 row weights (tid<16) and 16 col weights (tid>=16).
    if (tid < 32) {
        const bool isW = (tid >= 16);
        const int  p   = tid & 15;       // packed index: p = 2a + odd
        const int  a   = p >> 1;
        const int  odd = p & 1;
        const float lo  = rois[r * 4 + (isW ? 0 : 1)] * ROI_SCALE;   // x1 or y1
        const float hi  = rois[r * 4 + (isW ? 2 : 3)] * ROI_SCALE;   // x2 or y2
        float len = hi - lo + 1.0f;
        if (len < 0.0f) len = 0.0f;
        const float bin = len * (1.0f / 7.0f);       // /(A?-1), A?=8
        const float pos = lo + (float)a * bin;
        const int   lim = isW ? FW : FH;
        const bool  valid = (pos >= 0.0f) && (pos < (float)lim);
        float fs = floorf(pos);
        if (fs < 0.0f) fs = 0.0f;
        if (fs > (float)(lim - 2)) fs = (float)(lim - 2);
        const int   s    = (int)fs;
        const float frac = pos - fs;                 // frac computed from CLIPPED floor (matches ref)
        float wgt = odd ? frac : (1.0f - frac);
        if (!valid) wgt = 0.0f;                      // fold mask into weights
        wgt *= 0.5f;                                 // fold 2x2 avg-pool (0.5*0.5 = 0.25)
        if (isW) {
            sFc[p] = s + odd;
            if (a <= 6) sUw[a][p]     = wgt;         // Uw[j=a]
            if (a >= 1) sUw[a - 1][p] = wgt;         // Uw[j=a-1]
        } else {
            sFr[p] = s + odd;
            if (a <= 6) sUh[a][p]     = wgt;
            if (a >= 1) sUh[a - 1][p] = wgt;
        }
    }
    __syncthreads();

    const int lane = tid & 31;
    const int wv   = tid >> 5;
    const int half = lane >> 4;      // lane half selects K pair (A) / K rows (B)
    const int idx  = lane & 15;      // M (A-operand) or N (B-operand) index

    // ROI-constant WMMA operands + gather offsets (hoisted out of the channel loop).
    v2f aU[4];        // A slices of Uh for product 1
    v2f bW[4];        // B slices of Uw^T for product 2: B[k][j] = Uw[j][k]
    int goff0[4], goff1[4];
    #pragma unroll
    for (int kc = 0; kc < 4; ++kc) {
        const int k0 = 4 * kc + 2 * half;
        aU[kc][0] = sUh[idx][k0];
        aU[kc][1] = sUh[idx][k0 + 1];
        bW[kc][0] = sUw[idx][k0];
        bW[kc][1] = sUw[idx][k0 + 1];
        const int colb = sFc[idx];
        goff0[kc] = sFr[k0]     * FW + colb;
        goff1[kc] = sFr[k0 + 1] * FW + colb;
    }

    const int bid = bids[r];
    const float* base = feat + (size_t)bid * NCH * (FH * FW);
    float* outr = out + (size_t)r * NCH * 49;
    float* tbuf = &sT[wv][0];

    // Each wave handles 32 consecutive channels.
    for (int ci = 0; ci < 32; ++ci) {
        const int c = wv * 32 + ci;
        const float* fp = base + (size_t)c * (FH * FW);

        // Product 1: T = Uh * G  (G gathered on the fly into B-operand layout)
        v8f acc = {};
        #pragma unroll
        for (int kc = 0; kc < 4; ++kc) {
            v2f g;
            g[0] = fp[goff0[kc]];
            g[1] = fp[goff1[kc]];
            acc = __builtin_amdgcn_wmma_f32_16x16x4_f32(
                false, aU[kc], false, g, (short)0, acc, false, false);
        }

        // Bounce T (D layout) through LDS, lane-major: tbuf[lane*8 + row]
        {
            v4f* tv = (v4f*)&tbuf[lane * 8];
            tv[0] = (v4f){acc[0], acc[1], acc[2], acc[3]};
            tv[1] = (v4f){acc[4], acc[5], acc[6], acc[7]};
        }

        // Product 2: out = T * Uw^T ; re-read T in A-operand layout:
        // T[m][n] lives at tbuf[(n + 16*(m/8))*8 + (m%8)]
        v8f dacc = {};
        #pragma unroll
        for (int kc = 0; kc < 4; ++kc) {
            const int k0 = 4 * kc + 2 * half;
            v2f a2;
            a2[0] = tbuf[(k0     + 16 * (idx >> 3)) * 8 + (idx & 7)];
            a2[1] = tbuf[(k0 + 1 + 16 * (idx >> 3)) * 8 + (idx & 7)];
            dacc = __builtin_amdgcn_wmma_f32_16x16x4_f32(
                false, a2, false, bW[kc], (short)0, dacc, false, false);
        }

        // D layout: VGPR i, lanes 0-15 -> out[i][lane]. Valid region i<7, j<7.
        if (half == 0 && idx < 7) {
            float* op = outr + (size_t)c * 49 + idx;
            #pragma unroll
            for (int i = 0; i < 7; ++i)
                op[i * 7] = dacc[i];
        }
    }
}

extern "C" void kernel_launch(void* const* d_in, const int* in_sizes, int n_in,
                              void* d_out, int out_size, void* d_ws, size_t ws_size,
                              hipStream_t stream) {
    const float* feat = (const float*)d_in[0];   // (4, 256, 100, 152) f32
    const float* rois = (const float*)d_in[1];   // (R, 4) f32
    const int*   bids = (const int*)d_in[2];     // (R,) i32
    float* out = (float*)d_out;                  // (R, 256, 7, 7) f32
    const int R = in_sizes[2];

    roi_align_wmma_kernel<<<R, 256, 0, stream>>>(feat, rois, bids, out);
}